// PinnGenerator_80066780332271
// MI455X (gfx1250) — compile-verified
//
#include <hip/hip_runtime.h>

#ifndef __has_builtin
#define __has_builtin(x) 0
#endif

#define TX 64
#define TY 32
#define HALO_W (TX + 2)   /* 66 */
#define HALO_H (TY + 2)   /* 34 */
#define LP 68             /* LDS row pitch in floats */

#if defined(__HIP_DEVICE_COMPILE__) && __has_builtin(__builtin_amdgcn_global_load_async_to_lds_b32)
#define USE_ASYNC 1
#else
#define USE_ASYNC 0
#endif

typedef __attribute__((address_space(1))) int* gptr_i32;
typedef __attribute__((address_space(3))) int* lptr_i32;

__device__ __forceinline__ void wait_async0() {
#if __has_builtin(__builtin_amdgcn_s_wait_asynccnt)
  __builtin_amdgcn_s_wait_asynccnt(0);
#else
  asm volatile("s_wait_asynccnt 0" ::: "memory");
#endif
}

// One generalized Jacobi step:
//   yout = fc * f[1:-1,1:-1]  +  conv3x3(zero_pad(yin), C)
// where C = (k1+k2)/sum(k3), fc = h2 / (mu * sum(k3)).
// (Valid because kappa(1,mu) == mu exactly, so w is the constant field mu.)
// yin/yout: B x M x M (M=1022), f: B x N x N (N=1024).
__global__ __launch_bounds__(256) void pinn_jacobi_step(
    const float* __restrict__ yin, const float* __restrict__ f,
    const float* __restrict__ k1,  const float* __restrict__ k2,
    const float* __restrict__ k3,  const float* __restrict__ mu,
    float* __restrict__ yout, int M, int N, float h2)
{
  __shared__ float tile[HALO_H * LP];

  const int bx = blockIdx.x * TX;
  const int by = blockIdx.y * TY;
  const size_t plane = (size_t)M * (size_t)M;
  const float* yb = yin  + (size_t)blockIdx.z * plane;
  const float* fb = f    + (size_t)blockIdx.z * (size_t)N * (size_t)N;
  float*       ob = yout + (size_t)blockIdx.z * plane;

  // ---- Issue async staging of the 34x66 halo tile FIRST so the LDS-DMA
  //      overlaps the scalar coefficient setup below. ----
  const int tid = threadIdx.x;
  #pragma unroll
  for (int t = 0; t < (HALO_H * HALO_W + 255) / 256; ++t) {
    const int idx = tid + t * 256;
    if (idx < HALO_H * HALO_W) {
      const int r  = idx / HALO_W;
      const int cc = idx - r * HALO_W;
      const int gi = by + r - 1;
      const int gj = bx + cc - 1;
      const bool in = ((unsigned)gi < (unsigned)M) && ((unsigned)gj < (unsigned)M);
#if USE_ASYNC
      if (in) {
        __builtin_amdgcn_global_load_async_to_lds_b32(
            (gptr_i32)(yb + (size_t)gi * M + gj),
            (lptr_i32)&tile[r * LP + cc],
            0, 0);
      } else {
        tile[r * LP + cc] = 0.0f;   // zero Dirichlet halo
      }
#else
      tile[r * LP + cc] = in ? yb[(size_t)gi * M + gj] : 0.0f;
#endif
    }
  }

  // ---- Uniform coefficient setup (scalar loads + two divides), overlapped
  //      with the in-flight async tile copy. ----
  float s3 = 0.0f;
  #pragma unroll
  for (int t = 0; t < 9; ++t) s3 += k3[t];
  const float inv_s3 = 1.0f / s3;
  const float c0 = (k1[0] + k2[0]) * inv_s3;
  const float c1 = (k1[1] + k2[1]) * inv_s3;
  const float c2 = (k1[2] + k2[2]) * inv_s3;
  const float c3 = (k1[3] + k2[3]) * inv_s3;
  const float c4 = (k1[4] + k2[4]) * inv_s3;
  const float c5 = (k1[5] + k2[5]) * inv_s3;
  const float c6 = (k1[6] + k2[6]) * inv_s3;
  const float c7 = (k1[7] + k2[7]) * inv_s3;
  const float c8 = (k1[8] + k2[8]) * inv_s3;
  const float fc = (h2 * inv_s3) / mu[0];

#if USE_ASYNC
  wait_async0();
#endif
  __syncthreads();

  // ---- Compute: each thread sweeps 8 consecutive rows of one column,
  //      rolling a 3x3 register window (3 new LDS reads per point). ----
  const int tx = tid & (TX - 1);    // 0..63
  const int tq = tid >> 6;          // 0..3  -> rows [tq*8, tq*8+8)
  const int j  = bx + tx;
  const bool jok = (j < M);

  int lb = (tq * 8) * LP + tx;      // top-left of 3x3 window in halo coords
  float a0 = tile[lb], a1 = tile[lb + 1], a2 = tile[lb + 2];
  lb += LP;
  float b0 = tile[lb], b1 = tile[lb + 1], b2 = tile[lb + 2];

  #pragma unroll
  for (int rr = 0; rr < 8; ++rr) {
    lb += LP;
    const float d0 = tile[lb], d1 = tile[lb + 1], d2 = tile[lb + 2];
    const int i = by + tq * 8 + rr;
    if (jok && (i < M)) {
      float acc = fc * fb[(size_t)(i + 1) * N + (j + 1)];
      acc = __builtin_fmaf(c0, a0, acc);
      acc = __builtin_fmaf(c1, a1, acc);
      acc = __builtin_fmaf(c2, a2, acc);
      acc = __builtin_fmaf(c3, b0, acc);
      acc = __builtin_fmaf(c4, b1, acc);
      acc = __builtin_fmaf(c5, b2, acc);
      acc = __builtin_fmaf(c6, d0, acc);
      acc = __builtin_fmaf(c7, d1, acc);
      acc = __builtin_fmaf(c8, d2, acc);
      ob[(size_t)i * M + j] = acc;
    }
    a0 = b0; a1 = b1; a2 = b2;
    b0 = d0; b1 = d1; b2 = d2;
  }
}

extern "C" void kernel_launch(void* const* d_in, const int* in_sizes, int n_in,
                              void* d_out, int out_size, void* d_ws, size_t ws_size,
                              hipStream_t stream) {
  (void)n_in; (void)out_size; (void)ws_size;
  // setup_inputs order: x(unused), pre, f, mu, k1, k2, k3
  const float* pre = (const float*)d_in[1];
  const float* f   = (const float*)d_in[2];
  const float* mu  = (const float*)d_in[3];
  const float* k1  = (const float*)d_in[4];
  const float* k2  = (const float*)d_in[5];
  const float* k3  = (const float*)d_in[6];
  float* out = (float*)d_out;
  float* ws  = (float*)d_ws;

  const int N = 1024;
  const int M = N - 2;                       // 1022
  const int B = in_sizes[2] / (N * N);       // 8
  const float h = 1.0f / (float)(N - 1);
  const float h2 = h * h;

  dim3 grid((M + TX - 1) / TX, (M + TY - 1) / TY, B);  // 16 x 32 x 8
  dim3 block(256);

  // 11 stencil applications (1 explicit step + scan length 10).
  // Ping-pong: even k writes d_out, odd k writes d_ws; k=10 (last) -> d_out.
  const float* src = pre;
  for (int k = 0; k < 11; ++k) {
    float* dst = ((k & 1) == 0) ? out : ws;
    pinn_jacobi_step<<<grid, block, 0, stream>>>(src, f, k1, k2, k3, mu,
                                                 dst, M, N, h2);
    src = dst;
  }
}